// LegalExtractionModel_12962211299611
// MI455X (gfx1250) — compile-verified
//
#include <hip/hip_runtime.h>
#include <hip/hip_bf16.h>

typedef float v2f __attribute__((ext_vector_type(2)));
typedef float v8f __attribute__((ext_vector_type(8)));

#define N_NODES 30
#define DDIM    768
#define BATCH   256
#define SEQ     512
#define NPAIRS  435           // 30*29/2
#define NROWS   (BATCH * N_NODES)   // 7680 = 480 tiles of 16

// Kernel 1: S/T = E(7680x768) x [w1 w2] via V_WMMA_F32_16X16X4_F32.
// One wave per 16-row tile; K marched in steps of 4 (192 WMMA ops/wave).
__global__ void __launch_bounds__(256)
legal_matvec_wmma(const float* __restrict__ bert,
                  const float* __restrict__ re_w,
                  float* __restrict__ s_buf,
                  float* __restrict__ t_buf) {
    const int lane = threadIdx.x & 31;
    const int wave = threadIdx.x >> 5;
    const int tile = blockIdx.x * (blockDim.x >> 5) + wave;   // 0..479
    const int hi   = (lane >> 4) & 1;   // lanes 16-31 cover K+2,K+3
    const int lrow = lane & 15;         // A: row-in-tile / B: column N

    // A-matrix row pointer: global row -> (batch, node) inside bert_out
    const int row = tile * 16 + lrow;               // 0..7679
    const int b   = row / N_NODES;
    const int n   = row - b * N_NODES;
    const float* __restrict__ arow = bert + ((size_t)b * SEQ + n) * DDIM;

    // B-matrix column pointer: N=0 -> w1, N=1 -> w2.
    // Lanes with N>=2 read w1 (harmless: their C columns are never read),
    // keeping control flow uniform so EXEC is all-1s at the WMMA.
    const float* __restrict__ bcol = re_w + (lrow == 1 ? DDIM : 0);

    v8f c = {0.f, 0.f, 0.f, 0.f, 0.f, 0.f, 0.f, 0.f};

    #pragma unroll 8
    for (int k = 0; k < DDIM; k += 4) {
        const int off = k + 2 * hi;                 // this lane's K pair
        v2f a = *(const v2f*)(arow + off);          // A: K=off, off+1 of row M
        v2f bb = *(const v2f*)(bcol + off);         // B: K=off, off+1 of col N
        // (neg_a, A, neg_b, B, c_mod, C, reuse_a, reuse_b)
        c = __builtin_amdgcn_wmma_f32_16x16x4_f32(false, a, false, bb,
                                                  (short)0, c, false, false);
    }

    // C/D layout (16x16 f32): VGPR v -> (M = v + 8*hi, N = lane%16).
    // Column N=0 (lanes 0,16) = s; column N=1 (lanes 1,17) = t.
    if (lrow < 2) {
        float* __restrict__ dst = (lrow == 0) ? s_buf : t_buf;
        const int base = tile * 16 + hi * 8;
        #pragma unroll
        for (int v = 0; v < 8; ++v) dst[base + v] = c[v];
    }
}

// Kernel 2: per-batch pair scores. One block per batch; s/t staged in LDS.
__global__ void __launch_bounds__(256)
legal_pair_sigmoid(const float* __restrict__ s_buf,
                   const float* __restrict__ t_buf,
                   const float* __restrict__ re_b,
                   float* __restrict__ out) {
    __shared__ float sh[2 * N_NODES];
    const int b   = blockIdx.x;
    const int tid = threadIdx.x;

    if (tid < N_NODES)               sh[tid] = s_buf[b * N_NODES + tid];
    else if (tid < 2 * N_NODES)      sh[tid] = t_buf[b * N_NODES + (tid - N_NODES)];
    __syncthreads();

    const float bias = re_b[0];
    for (int p = tid; p < NPAIRS; p += blockDim.x) {
        // decode upper-triangle (k=1) pair index p -> (i, j)
        int i = 0, pp = p;
        while (pp >= (N_NODES - 1 - i)) { pp -= (N_NODES - 1 - i); ++i; }
        const int j = i + 1 + pp;
        const float x = sh[i] + sh[N_NODES + j] + bias;
        out[(size_t)b * NPAIRS + p] = 1.0f / (1.0f + __expf(-x));
    }
}

extern "C" void kernel_launch(void* const* d_in, const int* in_sizes, int n_in,
                              void* d_out, int out_size, void* d_ws, size_t ws_size,
                              hipStream_t stream) {
    const float* bert = (const float*)d_in[0];   // (256,512,768) f32
    const float* re_w = (const float*)d_in[1];   // (1536,) f32
    const float* re_b = (const float*)d_in[2];   // (1,) f32
    float* out = (float*)d_out;                  // 111360 f32

    float* s_buf = (float*)d_ws;                 // 7680 f32
    float* t_buf = s_buf + NROWS;                // 7680 f32

    // 480 tiles, 8 waves (256 threads) per block -> 60 blocks
    legal_matvec_wmma<<<NROWS / 16 / 8, 256, 0, stream>>>(bert, re_w, s_buf, t_buf);
    legal_pair_sigmoid<<<BATCH, 256, 0, stream>>>(s_buf, t_buf, re_b, out);
}